// MedianBlur_37160057045200
// MI455X (gfx1250) — compile-verified
//
#include <hip/hip_runtime.h>

// ---------------------------------------------------------------------------
// 7x7 median blur, exact (zero padding), f32, NCHW 8x3x512x512.
// Sorted-column + pruned Batcher-merge + staircase selection.
// ---------------------------------------------------------------------------

#ifndef USE_ASYNC
#if defined(__has_builtin)
#if __has_builtin(__builtin_amdgcn_global_load_async_to_lds_b32)
#define USE_ASYNC 1
#endif
#endif
#endif
#ifndef USE_ASYNC
#define USE_ASYNC 0
#endif

#define IMG_W 512
#define IMG_H 512
#define PLANES 24          // 8 batch * 3 channels
#define TILE_W 264         // staged input columns: [X0-4, X0+260)
#define LDS_W 268          // padded LDS row stride (floats), 16B-aligned rows

typedef __attribute__((address_space(1))) int* as1_i32p;
typedef __attribute__((address_space(3))) int* as3_i32p;

__device__ __forceinline__ void ce(float& a, float& b) {
  float t = fminf(a, b);
  b = fmaxf(a, b);
  a = t;
}

// Batcher odd-even merge: merges sorted halves a[LO..LO+N/2) and a[LO+N/2..LO+N).
// N power of two. Fully unrolled; all indices compile-time constants.
template <int LO, int N, int R, int LEN>
__device__ __forceinline__ void oem(float (&a)[LEN]) {
  if constexpr (2 * R < N) {
    oem<LO, N, 2 * R>(a);
    oem<LO + R, N, 2 * R>(a);
#pragma unroll
    for (int i = LO + R; i + R < LO + N; i += 2 * R) ce(a[i], a[i + R]);
  } else {
    ce(a[LO], a[LO + R]);
  }
}

// Batcher odd-even mergesort, N power of two.
template <int LO, int N, int LEN>
__device__ __forceinline__ void osort(float (&a)[LEN]) {
  if constexpr (N > 1) {
    osort<LO, N / 2>(a);
    osort<LO + N / 2, N / 2>(a);
    oem<LO, N, 1>(a);
  }
}

__global__ __launch_bounds__(256) void median7_kernel(const float* __restrict__ img,
                                                      float* __restrict__ out) {
  __shared__ __align__(16) float tile[14][LDS_W];

  const int X0 = blockIdx.x * 256;             // output tile x origin
  const int Y0 = blockIdx.y * 8;               // output tile y origin
  const size_t plane = (size_t)blockIdx.z * (IMG_W * IMG_H);
  const float* src = img + plane;
  const int tx = threadIdx.x;                  // 0..31, strip index
  const int ty = threadIdx.y;                  // 0..7, row in tile
  const int tid = ty * 32 + tx;

  // ---- stage halo tile (rows Y0-3..Y0+10, cols X0-4..X0+259) into LDS ----
  for (int r = 0; r < 14; ++r) {
    const int gy = Y0 - 3 + r;
    const bool yok = ((unsigned)gy < (unsigned)IMG_H);
    for (int c = tid; c < TILE_W; c += 256) {
      const int gx = X0 - 4 + c;
      const bool ok = yok && ((unsigned)gx < (unsigned)IMG_W);
#if USE_ASYNC
      if (ok) {
        __builtin_amdgcn_global_load_async_to_lds_b32(
            (as1_i32p)(src + gy * IMG_W + gx),
            (as3_i32p)&tile[r][c], 0, 0);
      } else {
        tile[r][c] = 0.0f;
      }
#else
      tile[r][c] = ok ? src[gy * IMG_W + gx] : 0.0f;
#endif
    }
  }
#if USE_ASYNC
#if defined(__has_builtin) && __has_builtin(__builtin_amdgcn_s_wait_asynccnt)
  __builtin_amdgcn_s_wait_asynccnt(0);
#else
  asm volatile("s_wait_asynccnt 0" ::: "memory");
#endif
#endif
  __syncthreads();

  const float FINF = __builtin_inff();

  // ---- per-thread strip: output pixels x = x0..x0+7 at row y = Y0+ty ----
  // LDS col index c maps to absolute column (X0-4)+c. Strip needs jr=1..14,
  // i.e. LDS cols tx*8+1 .. tx*8+14 (we load 16-wide aligned chunks).

  // Sorted columns col[jr][0..6] (+[7]=INF pad), jr in 1..14.
  float col[15][8];
  {
#pragma unroll
    for (int r = 0; r < 7; ++r) {
      float c16[16];
#pragma unroll
      for (int k = 0; k < 4; ++k) {
        const float4 q = *(const float4*)&tile[ty + r][tx * 8 + 4 * k];
        c16[4 * k + 0] = q.x;
        c16[4 * k + 1] = q.y;
        c16[4 * k + 2] = q.z;
        c16[4 * k + 3] = q.w;
      }
#pragma unroll
      for (int jr = 1; jr <= 14; ++jr) col[jr][r] = c16[jr];
    }
  }
#pragma unroll
  for (int jr = 1; jr <= 14; ++jr) {
    col[jr][7] = FINF;
    osort<0, 8>(col[jr]);
  }

  // Fixed even-aligned pair merges: P[pp] = merge(col[2pp+2], col[2pp+3]),
  // 14 real values sorted at [0..13], [14..15]=INF.
  float P[6][16];
#pragma unroll
  for (int pp = 0; pp < 6; ++pp) {
#pragma unroll
    for (int k = 0; k < 8; ++k) {
      P[pp][k] = col[2 * pp + 2][k];
      P[pp][8 + k] = col[2 * pp + 3][k];
    }
    oem<0, 16, 1>(P[pp]);
  }

  // Per pixel i: window = pairs {m+1,m+2,m+3} (P[m],P[m+1],P[m+2]) + lone col.
  float res[8];
#pragma unroll
  for (int i = 0; i < 8; ++i) {
    const int m = i >> 1;
    const int lone = (i & 1) ? (2 * m + 8) : (2 * m + 1);

    // N = merge(pairA, pairB): 28 real sorted at [0..27], 4 INF on top.
    float n32[32];
#pragma unroll
    for (int k = 0; k < 16; ++k) {
      n32[k] = P[m][k];
      n32[16 + k] = P[m + 1][k];
    }
    oem<0, 32, 1>(n32);
    // Median (rank 24 of 49) has N-rank in [3,24]; keep A = n32[3..24] (22),
    // 3 discarded-below are provably < median -> new target rank = 21.

    // R = merge(pairC, lone col): 21 real sorted at [0..20].
    float r32[32];
#pragma unroll
    for (int k = 0; k < 16; ++k) r32[k] = P[m + 2][k];
#pragma unroll
    for (int k = 0; k < 8; ++k) r32[16 + k] = col[lone][k];
#pragma unroll
    for (int k = 24; k < 32; ++k) r32[k] = FINF;
    oem<0, 32, 1>(r32);

    // Staircase selection: rank 21 (0-based) of A(22) ∪ B(21):
    //   ans = min over i=1..22 of max(A[i-1], B[21-i]), B[-1] = -inf case = A[21].
    float ans = n32[3 + 21];
#pragma unroll
    for (int t = 1; t <= 21; ++t)
      ans = fminf(ans, fmaxf(n32[3 + t - 1], r32[21 - t]));
    res[i] = ans;
  }

  // ---- store 8 results (two 16B stores) ----
  float* dst = out + plane + (size_t)(Y0 + ty) * IMG_W + (X0 + tx * 8);
  *(float4*)dst = make_float4(res[0], res[1], res[2], res[3]);
  *((float4*)dst + 1) = make_float4(res[4], res[5], res[6], res[7]);
}

extern "C" void kernel_launch(void* const* d_in, const int* in_sizes, int n_in,
                              void* d_out, int out_size, void* d_ws, size_t ws_size,
                              hipStream_t stream) {
  (void)in_sizes; (void)n_in; (void)out_size; (void)d_ws; (void)ws_size;
  const float* img = (const float*)d_in[0];
  float* out = (float*)d_out;
  dim3 grid(IMG_W / 256, IMG_H / 8, PLANES);   // (2, 64, 24)
  dim3 block(32, 8, 1);                        // 8 wave32 per block
  median7_kernel<<<grid, block, 0, stream>>>(img, out);
}